// multi_head_self_attention_37357625541231
// MI455X (gfx1250) — compile-verified
//
#include <hip/hip_runtime.h>
#include <hip/hip_bf16.h>

typedef __attribute__((ext_vector_type(2))) float v2f;
typedef __attribute__((ext_vector_type(4))) float f4;
typedef __attribute__((ext_vector_type(8))) float v8f;
typedef __attribute__((ext_vector_type(4))) unsigned int v4u;
typedef __attribute__((ext_vector_type(4))) int v4i;
typedef __attribute__((ext_vector_type(8))) int v8i;

#define SLEN  2048
#define EMB   1024
#define HD    64
#define NB    8
#define NHEAD 16

#if __has_builtin(__builtin_amdgcn_tensor_load_to_lds) && \
    __has_builtin(__builtin_amdgcn_s_wait_tensorcnt)
#define HAS_TDM 1
#else
#define HAS_TDM 0
#endif

static __device__ __forceinline__ v8f wmma4(v2f a, v2f b, v8f c) {
  // D = A(16x4 f32) * B(4x16 f32) + C(16x16 f32)
  return __builtin_amdgcn_wmma_f32_16x16x4_f32(false, a, false, b, (short)0, c,
                                               false, false);
}

// LDS byte offset of a __shared__ object (addrspace(3) ptr -> 32-bit offset)
static __device__ __forceinline__ unsigned lds_byte_off(const void* p) {
  return (unsigned)(unsigned long long)(
      __attribute__((address_space(3))) const void*)p;
}

#if HAS_TDM
// Issue a TDM 2D tile load Global->LDS per CDNA5 ISA section 8 (D# groups).
// tile_w/tensor_w/stride in elements (f32); pad codes per D# group1 encoding:
// pad_interval code c => pad after 2^(c+1) DWORDs; pad_amount code a => a+1 DWORDs.
static __device__ __forceinline__ void tdm_load_2d(
    unsigned lds, const void* gp, unsigned tile_w, unsigned tile_h,
    unsigned tensor_w, unsigned tensor_h, unsigned stride, unsigned padI,
    unsigned padA) {
  unsigned long long ga = (unsigned long long)gp;
  v4u g0;
  g0[0] = 1u;                                   // count=1, user descriptor
  g0[1] = lds;                                  // lds_addr (bytes)
  g0[2] = (unsigned)ga;                         // global_addr[31:0]
  g0[3] = (unsigned)((ga >> 32) & 0x01ffffffull) | 0x80000000u;  // [56:32] | type=2
  v8i g1;
  g1[0] = (int)((2u << 16)                      // data_size = 4 bytes
                | (1u << 20)                    // pad_enable
                | (padI << 22) | (padA << 25));
  g1[1] = (int)((tensor_w & 0xffffu) << 16);    // tensor_dim0[15:0]
  g1[2] = (int)(((tensor_w >> 16) & 0xffffu) | ((tensor_h & 0xffffu) << 16));
  g1[3] = (int)(((tensor_h >> 16) & 0xffffu) | ((tile_w & 0xffffu) << 16));
  g1[4] = (int)(tile_h & 0xffffu);              // tile_dim1; tile_dim2=0 (2D)
  g1[5] = (int)stride;                          // tensor_dim0_stride[31:0]
  g1[6] = 0;                                    // stride[47:32], dim1_stride lo
  g1[7] = 0;
  v4i gz = {0, 0, 0, 0};                        // groups 2/3 unused (2D)
  v8i gz8 = {0, 0, 0, 0, 0, 0, 0, 0};           // extra group (clang-23 form)
  __builtin_amdgcn_tensor_load_to_lds(g0, g1, gz, gz, gz8, 0);
}
#endif

// ---------------------------------------------------------------------------
// Kernel 1: fused QKV projection.  q/k/v[i,d] = sum_e x[i,e] * W[d,e]
// Block: 128 rows of x (8 waves x 16 rows), N = 3*64.  K streamed in chunks
// of 32 with TDM double buffering (LDS row stride 36 = 32 + 4 pad).
// ---------------------------------------------------------------------------
__global__ __launch_bounds__(256) void qkv_proj_kernel(
    const float* __restrict__ x, const float* __restrict__ Wq,
    const float* __restrict__ Wk, const float* __restrict__ Wv,
    float* __restrict__ q, float* __restrict__ k, float* __restrict__ v) {
  __shared__ float Xs[2][128][36];
  __shared__ float Ws[2][3][64][36];
  const int tid  = threadIdx.x;
  const int wv   = tid >> 5;
  const int lane = tid & 31;
  const int half = lane >> 4;
  const int ln   = lane & 15;
  const int r0   = blockIdx.x * 128;

  v8f acc[3][4];
#pragma unroll
  for (int g = 0; g < 3; ++g)
#pragma unroll
    for (int t = 0; t < 4; ++t) acc[g][t] = {};

#if HAS_TDM
  if (wv == 0) {  // prefetch stage 0
    tdm_load_2d(lds_byte_off(&Xs[0][0][0]), x + (size_t)r0 * EMB, 32, 128, EMB,
                NB * SLEN, EMB, 4, 3);
    tdm_load_2d(lds_byte_off(&Ws[0][0][0][0]), Wq, 32, 64, EMB, HD, EMB, 4, 3);
    tdm_load_2d(lds_byte_off(&Ws[0][1][0][0]), Wk, 32, 64, EMB, HD, EMB, 4, 3);
    tdm_load_2d(lds_byte_off(&Ws[0][2][0][0]), Wv, 32, 64, EMB, HD, EMB, 4, 3);
  }
#endif

  for (int ch = 0; ch < EMB / 32; ++ch) {
    const int buf = ch & 1;
#if HAS_TDM
    if (wv == 0) {
      if (ch + 1 < EMB / 32) {  // issue next stage into other buffer
        const int e0 = (ch + 1) * 32, nb = buf ^ 1;
        tdm_load_2d(lds_byte_off(&Xs[nb][0][0]),
                    x + (size_t)r0 * EMB + e0, 32, 128, EMB, NB * SLEN, EMB, 4, 3);
        tdm_load_2d(lds_byte_off(&Ws[nb][0][0][0]), Wq + e0, 32, 64, EMB, HD, EMB, 4, 3);
        tdm_load_2d(lds_byte_off(&Ws[nb][1][0][0]), Wk + e0, 32, 64, EMB, HD, EMB, 4, 3);
        tdm_load_2d(lds_byte_off(&Ws[nb][2][0][0]), Wv + e0, 32, 64, EMB, HD, EMB, 4, 3);
        __builtin_amdgcn_s_wait_tensorcnt(4);  // stage ch complete, next in flight
      } else {
        __builtin_amdgcn_s_wait_tensorcnt(0);  // final stage complete
      }
    }
    __syncthreads();
#else
    {  // synchronous fallback staging
      const int e0 = ch * 32;
#pragma unroll
      for (int t = 0; t < 4; ++t) {
        int f = tid + t * 256, row = f >> 3, c4 = f & 7;
        *(f4*)&Xs[buf][row][c4 * 4] =
            *(const f4*)(x + (size_t)(r0 + row) * EMB + e0 + c4 * 4);
      }
#pragma unroll
      for (int t = 0; t < 6; ++t) {
        int f = tid + t * 256, g = f >> 9, rem = f & 511;
        int d = rem >> 3, c4 = rem & 7;
        const float* wp = (g == 0) ? Wq : ((g == 1) ? Wk : Wv);
        *(f4*)&Ws[buf][g][d][c4 * 4] =
            *(const f4*)(wp + (size_t)d * EMB + e0 + c4 * 4);
      }
    }
    __syncthreads();
#endif

#pragma unroll
    for (int e4 = 0; e4 < 8; ++e4) {
      const int ec = e4 * 4 + 2 * half;
      v2f a = *(const v2f*)&Xs[buf][wv * 16 + ln][ec];
#pragma unroll
      for (int g = 0; g < 3; ++g) {
#pragma unroll
        for (int t = 0; t < 4; ++t) {
          v2f b = *(const v2f*)&Ws[buf][g][t * 16 + ln][ec];
          acc[g][t] = wmma4(a, b, acc[g][t]);
        }
      }
    }
    __syncthreads();
  }

  const int rw = r0 + wv * 16;
#pragma unroll
  for (int g = 0; g < 3; ++g) {
    float* op = (g == 0) ? q : ((g == 1) ? k : v);
#pragma unroll
    for (int t = 0; t < 4; ++t)
#pragma unroll
      for (int j = 0; j < 8; ++j)
        op[(size_t)(rw + j + 8 * half) * HD + t * 16 + ln] = acc[g][t][j];
  }
}

// ---------------------------------------------------------------------------
// Kernel 2: per-(batch,key) softmax stats over the QUERY axis.
// Block: one batch, 128 keys (8 waves x 16).  Streams all 2048 queries in
// TDM double-buffered tiles of 64 rows.  Stores (m, 1/l) as float2.
// ---------------------------------------------------------------------------
__global__ __launch_bounds__(256) void stats_kernel(
    const float* __restrict__ q, const float* __restrict__ kmat,
    float2* __restrict__ stats) {
  __shared__ float Qs[2][64][68];
  const int tid  = threadIdx.x;
  const int wv   = tid >> 5;
  const int lane = tid & 31;
  const int half = lane >> 4;
  const int ln   = lane & 15;
  const int b    = blockIdx.x >> 4;
  const int kblk = blockIdx.x & 15;
  const int keyBase = kblk * 128 + wv * 16;
  const float* kb = kmat + (size_t)b * SLEN * HD;
  const float* qb = q + (size_t)b * SLEN * HD;

  // K-matrix rows as resident WMMA A fragments (M = key, K = head dim)
  v2f ak[16];
#pragma unroll
  for (int e4 = 0; e4 < 16; ++e4)
    ak[e4] = *(const v2f*)(kb + (size_t)(keyBase + ln) * HD + e4 * 4 + 2 * half);

  float mrun[8], lrun[8];
#pragma unroll
  for (int j = 0; j < 8; ++j) {
    mrun[j] = -__builtin_inff();
    lrun[j] = 0.0f;
  }

#if HAS_TDM
  if (wv == 0)
    tdm_load_2d(lds_byte_off(&Qs[0][0][0]), qb, 64, 64, HD, SLEN, HD, 5, 3);
#endif

  for (int ch = 0; ch < SLEN / 64; ++ch) {
    const int buf = ch & 1;
#if HAS_TDM
    if (wv == 0) {
      if (ch + 1 < SLEN / 64) {
        tdm_load_2d(lds_byte_off(&Qs[buf ^ 1][0][0]),
                    qb + (size_t)(ch + 1) * 64 * HD, 64, 64, HD, SLEN, HD, 5, 3);
        __builtin_amdgcn_s_wait_tensorcnt(1);
      } else {
        __builtin_amdgcn_s_wait_tensorcnt(0);
      }
    }
    __syncthreads();
#else
    {
      const int i0 = ch * 64;
#pragma unroll
      for (int t = 0; t < 4; ++t) {
        int f = tid + t * 256, row = f >> 4, c4 = f & 15;
        *(f4*)&Qs[buf][row][c4 * 4] =
            *(const f4*)(qb + (size_t)(i0 + row) * HD + c4 * 4);
      }
    }
    __syncthreads();
#endif
#pragma unroll
    for (int s = 0; s < 4; ++s) {
      v8f c = {};
#pragma unroll
      for (int e4 = 0; e4 < 16; ++e4) {
        v2f bq = *(const v2f*)&Qs[buf][s * 16 + ln][e4 * 4 + 2 * half];
        c = wmma4(ak[e4], bq, c);  // C: rows = keys, cols = queries
      }
      // per-lane online softmax over this lane's query subsequence
#pragma unroll
      for (int j = 0; j < 8; ++j) {
        float vv = c[j];
        float mn = fmaxf(mrun[j], vv);
        lrun[j] = lrun[j] * __expf(mrun[j] - mn) + __expf(vv - mn);
        mrun[j] = mn;
      }
    }
    __syncthreads();
  }

  // merge the 16 lanes of each half (disjoint query subsets)
#pragma unroll
  for (int j = 0; j < 8; ++j) {
#pragma unroll
    for (int mk = 1; mk < 16; mk <<= 1) {
      float om = __shfl_xor(mrun[j], mk, 32);
      float ol = __shfl_xor(lrun[j], mk, 32);
      float mn = fmaxf(mrun[j], om);
      lrun[j] = lrun[j] * __expf(mrun[j] - mn) + ol * __expf(om - mn);
      mrun[j] = mn;
    }
  }
#pragma unroll
  for (int r = 0; r < 8; ++r) {
    if (ln == r) {
      int key = keyBase + r + 8 * half;
      stats[(size_t)b * SLEN + key] = make_float2(mrun[r], 1.0f / lrun[r]);
    }
  }
}

// ---------------------------------------------------------------------------
// Kernel 3: out[b,i,d] = sum_k exp(s[b,i,k]-m[b,k]) * invl[b,k] * v[b,k,d]
// Block: one batch, 128 queries (8 waves x 16).  K/V tiles of 16 keys are
// TDM double-buffered; scores recomputed with WMMA; P transposed through a
// per-wave LDS tile; P*V with WMMA.  Output tiled x16 across heads.
// ---------------------------------------------------------------------------
__global__ __launch_bounds__(256) void attn_out_kernel(
    const float* __restrict__ q, const float* __restrict__ kmat,
    const float* __restrict__ vmat, const float2* __restrict__ stats,
    float* __restrict__ out) {
  __shared__ float Ks[2][16][68];
  __shared__ float Vs[2][16][68];
  __shared__ float Ps[8][16][18];  // per-wave P transpose tile
  __shared__ float2 Ss[SLEN];
  const int tid  = threadIdx.x;
  const int wv   = tid >> 5;
  const int lane = tid & 31;
  const int half = lane >> 4;
  const int ln   = lane & 15;
  const int b    = blockIdx.x >> 4;
  const int iblk = blockIdx.x & 15;
  const int iBase = iblk * 128 + wv * 16;
  const float* qb = q + (size_t)b * SLEN * HD;
  const float* kb = kmat + (size_t)b * SLEN * HD;
  const float* vb = vmat + (size_t)b * SLEN * HD;

#pragma unroll
  for (int t = 0; t < 8; ++t)
    Ss[tid + t * 256] = stats[(size_t)b * SLEN + tid + t * 256];

  // Q rows as resident A fragments
  v2f aq[16];
#pragma unroll
  for (int e4 = 0; e4 < 16; ++e4)
    aq[e4] = *(const v2f*)(qb + (size_t)(iBase + ln) * HD + e4 * 4 + 2 * half);

  v8f acc[4];
#pragma unroll
  for (int t = 0; t < 4; ++t) acc[t] = {};

#if HAS_TDM
  if (wv == 0) {
    tdm_load_2d(lds_byte_off(&Ks[0][0][0]), kb, 64, 16, HD, SLEN, HD, 5, 3);
    tdm_load_2d(lds_byte_off(&Vs[0][0][0]), vb, 64, 16, HD, SLEN, HD, 5, 3);
  }
#endif
  __syncthreads();

  for (int k0 = 0; k0 < SLEN; k0 += 16) {
    const int buf = (k0 >> 4) & 1;
#if HAS_TDM
    if (wv == 0) {
      if (k0 + 16 < SLEN) {
        const int nb = buf ^ 1;
        tdm_load_2d(lds_byte_off(&Ks[nb][0][0]), kb + (size_t)(k0 + 16) * HD,
                    64, 16, HD, SLEN, HD, 5, 3);
        tdm_load_2d(lds_byte_off(&Vs[nb][0][0]), vb + (size_t)(k0 + 16) * HD,
                    64, 16, HD, SLEN, HD, 5, 3);
        __builtin_amdgcn_s_wait_tensorcnt(2);
      } else {
        __builtin_amdgcn_s_wait_tensorcnt(0);
      }
    }
    __syncthreads();
#else
    {
      int row = tid >> 4, c4 = tid & 15;
      *(f4*)&Ks[buf][row][c4 * 4] =
          *(const f4*)(kb + (size_t)(k0 + row) * HD + c4 * 4);
      *(f4*)&Vs[buf][row][c4 * 4] =
          *(const f4*)(vb + (size_t)(k0 + row) * HD + c4 * 4);
    }
    __syncthreads();
#endif

    // scores tile: rows = queries, cols = keys
    v8f c = {};
#pragma unroll
    for (int e4 = 0; e4 < 16; ++e4) {
      v2f bk = *(const v2f*)&Ks[buf][ln][e4 * 4 + 2 * half];
      c = wmma4(aq[e4], bk, c);
    }
    float2 st = Ss[k0 + ln];  // column (key) stats for this lane
#pragma unroll
    for (int j = 0; j < 8; ++j) c[j] = __expf(c[j] - st.x) * st.y;

    // transpose P (C-layout -> A-layout) through per-wave LDS tile
#pragma unroll
    for (int j = 0; j < 8; ++j) Ps[wv][j + 8 * half][ln] = c[j];
    asm volatile("s_wait_dscnt 0" ::: "memory");  // DS in-order; keep compiler honest

#pragma unroll
    for (int c4 = 0; c4 < 4; ++c4) {
      v2f ap = *(const v2f*)&Ps[wv][ln][c4 * 4 + 2 * half];
#pragma unroll
      for (int t = 0; t < 4; ++t) {
        v2f bvv;
        bvv[0] = Vs[buf][c4 * 4 + 2 * half + 0][t * 16 + ln];
        bvv[1] = Vs[buf][c4 * 4 + 2 * half + 1][t * 16 + ln];
        acc[t] = wmma4(ap, bvv, acc[t]);
      }
    }
    __syncthreads();
  }

  // write out, replicated across the 16 identical heads
#pragma unroll
  for (int t = 0; t < 4; ++t)
#pragma unroll
    for (int j = 0; j < 8; ++j) {
      float val = acc[t][j];
      size_t rowoff = ((size_t)b * SLEN + iBase + j + 8 * half) * (size_t)EMB;
#pragma unroll
      for (int h = 0; h < NHEAD; ++h)
        out[rowoff + h * HD + t * 16 + ln] = val;
    }
}

// ---------------------------------------------------------------------------
extern "C" void kernel_launch(void* const* d_in, const int* in_sizes, int n_in,
                              void* d_out, int out_size, void* d_ws,
                              size_t ws_size, hipStream_t stream) {
  const float* x  = (const float*)d_in[0];
  const float* Wq = (const float*)d_in[1];
  const float* Wk = (const float*)d_in[2];
  const float* Wv = (const float*)d_in[3];
  float* out = (float*)d_out;

  const size_t QN = (size_t)NB * SLEN * HD;  // 1M floats each
  float* q = (float*)d_ws;
  float* k = q + QN;
  float* v = k + QN;
  float2* stats = (float2*)(v + QN);  // NB*SLEN float2

  qkv_proj_kernel<<<(NB * SLEN) / 128, 256, 0, stream>>>(x, Wq, Wk, Wv, q, k, v);
  stats_kernel<<<NB * (SLEN / 128), 256, 0, stream>>>(q, k, stats);
  attn_out_kernel<<<NB * (SLEN / 128), 256, 0, stream>>>(q, k, v, stats, out);
}